// S4D_68436008894952
// MI455X (gfx1250) — compile-verified
//
#include <hip/hip_runtime.h>
#include <hip/hip_bf16.h>

// S4D on MI455X (gfx1250, wave32).
// Kernel 1: per-channel complex Vandermonde -> K[c, l] (workspace, 8MB).
// Kernel 2: causal conv as per-channel Toeplitz matmul via V_WMMA_F32_16X16X4_F32,
//           double-buffered GLOBAL_LOAD_ASYNC_TO_LDS_B128 staging of U.
//
// The chunk loop is fully unrolled so the diagonal-tile index q is a
// compile-time constant: the active-tile set per chunk (jj in [q+1,15]) has
// compile-time bounds (no runtime guards in the k-loop), acc[] indices are
// constants (registers), and the B fragment is loaded once per k-step and
// shared by all active tiles of the wave.

#define NCH   512
#define LEN   4096
#define NB    16          // batch == WMMA N dimension
#define NST   32          // D_STATE/2 complex modes
#define CHUNK 256         // s-chunk staged in LDS
#define UPAD  260         // padded row stride: 16B-aligned rows, bank offset 4/row
#define KPAD  20          // zero pad covering max diagonal overshoot (<=15)
#define NTHREADS 512      // 16 waves of 32

#if defined(__has_builtin)
#if __has_builtin(__builtin_amdgcn_global_load_async_to_lds_b128) && \
    __has_builtin(__builtin_amdgcn_s_wait_asynccnt)
#define USE_ASYNC_LDS 1
#endif
#endif

typedef float v2f __attribute__((ext_vector_type(2)));
typedef float v8f __attribute__((ext_vector_type(8)));
typedef int   v4i __attribute__((vector_size(16)));      // b128 payload type
typedef v4i __attribute__((address_space(1))) v4i_g;     // global
typedef v4i __attribute__((address_space(3))) v4i_l;     // LDS (32-bit ptr)

// ---------------------------------------------------------------------------
// Kernel 1: K[c,l] = 2 * sum_n Re( (B*C)[c,n] * dA[c,n]^l )
// One wave per channel; lane n owns mode n; sequential power recurrence.
// ---------------------------------------------------------------------------
__global__ __launch_bounds__(32)
void s4d_kernel_K(const float* __restrict__ dt_log,
                  const float* __restrict__ logA,
                  const float* __restrict__ Aim,
                  const float* __restrict__ Cr,
                  const float* __restrict__ Ci,
                  const float* __restrict__ Br,
                  const float* __restrict__ Bi,
                  float* __restrict__ Kout) {
    const int c = blockIdx.x;
    const int n = threadIdx.x;            // 0..31 (wave32 lane == mode)
    const int idx = c * NST + n;

    const float dt = __expf(dt_log[0]);
    const float ar = -__expf(logA[idx]);
    const float ai = Aim[idx];

    // z = dt*A/2 ; dA = (1+z)/(1-z)
    const float zr = 0.5f * dt * ar, zi = 0.5f * dt * ai;
    const float dr = 1.0f - zr, di = -zi;
    const float inv_den = 1.0f / (dr * dr + di * di);
    const float nr = 1.0f + zr, ni = zi;
    const float dAr = (nr * dr + ni * di) * inv_den;
    const float dAi = (ni * dr - nr * di) * inv_den;

    // BC = B * C (complex)
    const float br = Br[idx], bi = Bi[idx], cr = Cr[idx], ci = Ci[idx];
    const float bcr = br * cr - bi * ci;
    const float bci = br * ci + bi * cr;

    float pr = 1.0f, pi = 0.0f;           // dA^0
    float* krow = Kout + (size_t)c * LEN;
    for (int l = 0; l < LEN; ++l) {
        float contrib = bcr * pr - bci * pi;   // Re(BC * dA^l)
        #pragma unroll
        for (int off = 16; off > 0; off >>= 1)
            contrib += __shfl_xor(contrib, off, 32);
        if (n == 0) krow[l] = 2.0f * contrib;
        const float npr = pr * dAr - pi * dAi;
        pi = pr * dAi + pi * dAr;
        pr = npr;
    }
}

// ---------------------------------------------------------------------------
// Kernel 2: y[b,c,t] = sum_{s<=t} K[c,t-s] * U[b,c,s]
// One workgroup per channel; 16 waves. Wave w owns 16 persistent 16x16
// (time x batch) accumulator tiles: t0 = 256*jj + 16*w.
// ---------------------------------------------------------------------------
__global__ __launch_bounds__(NTHREADS)
void s4d_conv(const float* __restrict__ U,
              const float* __restrict__ Kker,
              float* __restrict__ Y) {
    __shared__ float Krev[LEN + KPAD];         // reversed kernel + zero pad
    __shared__ float Ush[2 * NB * UPAD];       // double-buffered U chunk: ~33KB

    const int c    = blockIdx.x;
    const int tid  = threadIdx.x;
    // Force wave id into an SGPR: all tile bounds become scalar -> WMMA runs
    // with EXEC all-ones and loop control is s_cbranch, not exec masking.
    const int w    = __builtin_amdgcn_readfirstlane(tid >> 5);
    const int lane = tid & 31;
    const int m    = lane & 15;           // A row within tile
    const int nb   = lane & 15;           // B column (batch)
    const int khalf = (lane >> 4) * 2;    // lanes 0-15 -> k=0,1 ; 16-31 -> k=2,3
    const int hi    = (lane >> 4) * 8;    // D: lanes 16-31 hold M = v+8
    const size_t CL = (size_t)NCH * LEN;
    const float* urow = U + (size_t)c * LEN;
    const float* krow = Kker + (size_t)c * LEN;

    // stage K reversed into LDS, zero pad the tail:
    //   Krev[j] = K[LEN-1-j] (j < LEN), Krev[LEN..] = 0
    for (int j = tid; j < LEN + KPAD; j += NTHREADS)
        Krev[j] = (j < LEN) ? krow[LEN - 1 - j] : 0.0f;

    // stage one 16(batch) x 256(s) chunk of U into LDS buffer at `bufbase`
    auto stage_chunk = [&](int s0c, int bufbase) {
#ifdef USE_ASYNC_LDS
        // 1024 x b128 async copies, 2 per thread; rows are 16B aligned (UPAD=260)
        for (int i = tid; i < NB * (CHUNK / 4); i += NTHREADS) {
            const int b  = i >> 6;            // batch row
            const int s4 = i & 63;            // float4 index within row
            const float* g = urow + (size_t)b * CL + s0c + s4 * 4;
            v4i_g* gsrc = (v4i_g*)(uintptr_t)g;
            v4i_l* ldst = (v4i_l*)(uint32_t)(uintptr_t)&Ush[bufbase + b * UPAD + s4 * 4];
            __builtin_amdgcn_global_load_async_to_lds_b128(gsrc, ldst, 0, 0);
        }
#else
        for (int i = tid; i < NB * CHUNK; i += NTHREADS) {
            const int b = i >> 8;
            const int s = i & (CHUNK - 1);
            Ush[bufbase + b * UPAD + s] = urow[(size_t)b * CL + s0c + s];
        }
#endif
    };

    v8f acc[16];
    #pragma unroll
    for (int jj = 0; jj < 16; ++jj) acc[jj] = (v8f){0,0,0,0,0,0,0,0};

    // per-lane invariant part of the reversed-Toeplitz A index:
    //   Krev index = (LEN-1 - m + khalf) + (s0 - t0) + sb   (pair ascending)
    const int aoff = (LEN - 1) - m + khalf;

    stage_chunk(0, 0);                         // prologue: fill buffer 0

    // Fully unrolled chunk loop: q is a compile-time constant everywhere.
    #pragma unroll
    for (int q = 0; q < 16; ++q) {
        const int s0 = q * CHUNK;
#ifdef USE_ASYNC_LDS
        __builtin_amdgcn_s_wait_asynccnt(0);   // my async writes have landed
#endif
        __syncthreads();                       // chunk visible + other buf free
        if (q < 15) stage_chunk(s0 + CHUNK, ((q & 1) ^ 1) * NB * UPAD);

        const float* Ubase = &Ush[(q & 1) * NB * UPAD + nb * UPAD + khalf];

        // (a) full tiles jj in [q+1, 15]: compile-time bounds, B fragment
        //     loaded once per k-step and shared by all active tiles.
        if (q < 15) {
            #pragma unroll 2
            for (int sb = 0; sb < CHUNK; sb += 4) {
                v2f bf;
                bf.x = Ubase[sb];
                bf.y = Ubase[sb + 1];
                #pragma unroll
                for (int jj = q + 1; jj < 16; ++jj) {
                    const float* Abase = &Krev[aoff - 16 * w + s0 - 256 * jj];
                    v2f a;
                    a.x = Abase[sb];
                    a.y = Abase[sb + 1];
                    acc[jj] = __builtin_amdgcn_wmma_f32_16x16x4_f32(
                        false, a, false, bf, (short)0, acc[jj], false, false);
                }
            }
        }

        // (b) the diagonal tile jj == q (constant acc index): partial k-range
        //     16w+16; the zero pad of Krev supplies the masked corner.
        {
            const float* Abase = &Krev[aoff - 16 * w];   // s0 - t0 == -16w here
            const int send = 16 * w + 16;                // scalar (per-wave)
            #pragma unroll 4
            for (int sb = 0; sb < send; sb += 4) {
                v2f a, bf;
                a.x  = Abase[sb];
                a.y  = Abase[sb + 1];
                bf.x = Ubase[sb];
                bf.y = Ubase[sb + 1];
                acc[q] = __builtin_amdgcn_wmma_f32_16x16x4_f32(
                    false, a, false, bf, (short)0, acc[q], false, false);
            }
        }
    }

    // Output: stage each 256-wide time band through LDS for coalesced,
    // non-temporal stores. D layout: vgpr v -> (t = t0 + v + hi, b = lane&15).
    #pragma unroll 1
    for (int jj = 0; jj < 16; ++jj) {
        __syncthreads();
        const int tloc0 = 16 * w;
        #pragma unroll
        for (int v = 0; v < 8; ++v)
            Ush[nb * UPAD + tloc0 + v + hi] = acc[jj][v];
        __syncthreads();
        for (int i = tid; i < NB * CHUNK; i += NTHREADS) {
            const int b  = i >> 8;
            const int tl = i & (CHUNK - 1);
            __builtin_nontemporal_store(
                Ush[b * UPAD + tl],
                &Y[(size_t)b * CL + (size_t)c * LEN + 256 * jj + tl]);
        }
    }
}

// ---------------------------------------------------------------------------
extern "C" void kernel_launch(void* const* d_in, const int* in_sizes, int n_in,
                              void* d_out, int out_size, void* d_ws, size_t ws_size,
                              hipStream_t stream) {
    const float* U      = (const float*)d_in[0];
    const float* dt_log = (const float*)d_in[1];
    const float* logA   = (const float*)d_in[2];
    const float* Aim    = (const float*)d_in[3];
    const float* Cr     = (const float*)d_in[4];
    const float* Ci     = (const float*)d_in[5];
    const float* Br     = (const float*)d_in[6];
    const float* Bi     = (const float*)d_in[7];
    float* Y    = (float*)d_out;
    float* Kbuf = (float*)d_ws;                 // 512*4096*4 = 8MB scratch

    s4d_kernel_K<<<NCH, 32, 0, stream>>>(dt_log, logA, Aim, Cr, Ci, Br, Bi, Kbuf);
    s4d_conv<<<NCH, NTHREADS, 0, stream>>>(U, Kbuf, Y);
}